// Actor1_48464410968726
// MI455X (gfx1250) — compile-verified
//
#include <hip/hip_runtime.h>
#include <hip/hip_bf16.h>
#include <math.h>

#define NXC 384
#define NYC 384
#define NROW 768
#define DATA_C 4096
#define DM 128
#define HIDC 128
#define NPAIR ((NXC*(NXC-1))/2)   /* 73536 */
#define EPSC 1e-5f
#define ALPHAC 0.0f

typedef float v2f __attribute__((ext_vector_type(2)));
typedef float v8f __attribute__((ext_vector_type(8)));

union V8U { v8f v; float f[8]; };

static __device__ __forceinline__ v8f wmma_f32x4(v2f a, v2f b, v8f c) {
  // D = A(16x4,f32) x B(4x16,f32) + C(16x16,f32); exact fp32
  return __builtin_amdgcn_wmma_f32_16x16x4_f32(false, a, false, b, (short)0, c, false, false);
}

// monotone float<->uint encoding for atomicMax over signed floats
static __device__ __forceinline__ unsigned encf(float f) {
  unsigned u = __float_as_uint(f);
  return (u & 0x80000000u) ? ~u : (u | 0x80000000u);
}
static __device__ __forceinline__ float decf(unsigned e) {
  return (e & 0x80000000u) ? __uint_as_float(e ^ 0x80000000u) : __uint_as_float(~e);
}

// ---------------- LayerNorm row statistics ----------------
__global__ void k_stats(const float* __restrict__ X, const float* __restrict__ Y,
                        float* __restrict__ mu, float* __restrict__ rstd) {
  int row = blockIdx.x;
  const float* src = (row < NXC) ? (X + (size_t)row * DATA_C)
                                 : (Y + (size_t)(row - NXC) * DATA_C);
  __shared__ float s_s[256], s_q[256];
  float s = 0.f, q = 0.f;
  for (int d = threadIdx.x; d < DATA_C; d += 256) { float v = src[d]; s += v; q += v * v; }
  s_s[threadIdx.x] = s; s_q[threadIdx.x] = q; __syncthreads();
  for (int o = 128; o > 0; o >>= 1) {
    if (threadIdx.x < o) { s_s[threadIdx.x] += s_s[threadIdx.x + o]; s_q[threadIdx.x] += s_q[threadIdx.x + o]; }
    __syncthreads();
  }
  if (threadIdx.x == 0) {
    float m = s_s[0] * (1.0f / DATA_C);
    float var = s_q[0] * (1.0f / DATA_C) - m * m;
    mu[row] = m; rstd[row] = rsqrtf(var + EPSC);
  }
}

// ---------------- Fused LN + reduce GEMM: E[768,128] = LN(X||Y) @ red_W^T + red_b, nan0 ----------------
// 48 m-tiles x 8 n-tiles = 384 waves; wave-per-16x16-tile, K stepped by 4 with f32 WMMA.
__global__ void k_ln_gemm(const float* __restrict__ X, const float* __restrict__ Y,
                          const float* __restrict__ mu, const float* __restrict__ rstd,
                          const float* __restrict__ g, const float* __restrict__ bt,
                          const float* __restrict__ W, const float* __restrict__ bias,
                          float* __restrict__ E) {
  int wave = (blockIdx.x * blockDim.x + threadIdx.x) >> 5;   // 0..383
  int lane = threadIdx.x & 31, half = lane >> 4, lm = lane & 15;
  int mt = wave >> 3, nt = wave & 7;
  int row = mt * 16 + lm;
  const float* arow = (row < NXC) ? (X + (size_t)row * DATA_C)
                                  : (Y + (size_t)(row - NXC) * DATA_C);
  float m = mu[row], rs = rstd[row];
  const float* brow = W + (size_t)(nt * 16 + lm) * DATA_C;
  v8f acc = {};
  int koff = half * 2;
  #pragma unroll 4
  for (int k = 0; k < DATA_C; k += 4) {
    int ka = k + koff;
    v2f a, b;
    a.x = (arow[ka]     - m) * rs * g[ka]     + bt[ka];
    a.y = (arow[ka + 1] - m) * rs * g[ka + 1] + bt[ka + 1];
    b.x = brow[ka]; b.y = brow[ka + 1];
    acc = wmma_f32x4(a, b, acc);
  }
  V8U u; u.v = acc;
  int col = nt * 16 + lm;
  float bv = bias[col];
  #pragma unroll
  for (int v = 0; v < 8; ++v) {
    float val = u.f[v] + bv;
    if (val != val) val = 0.0f;                 // _nan0
    int r = mt * 16 + v + half * 8;
    E[(size_t)r * DM + col] = val;
  }
}

// ---------------- Generic small WMMA GEMM: C[384,384] = (A ∘ scale) @ B^T, K=128 ----------------
__global__ void k_gram(const float* __restrict__ A, const float* __restrict__ B,
                       const float* __restrict__ scale, float* __restrict__ C) {
  int wave = (blockIdx.x * blockDim.x + threadIdx.x) >> 5;   // 0..575
  int lane = threadIdx.x & 31, half = lane >> 4, lm = lane & 15;
  int mt = wave / 24, nt = wave % 24;
  const float* arow = A + (size_t)(mt * 16 + lm) * DM;
  const float* brow = B + (size_t)(nt * 16 + lm) * DM;
  v8f acc = {};
  int koff = half * 2;
  #pragma unroll 8
  for (int k = 0; k < DM; k += 4) {
    int ka = k + koff;
    v2f a, b;
    float s0 = scale ? scale[ka] : 1.0f;
    float s1 = scale ? scale[ka + 1] : 1.0f;
    a.x = arow[ka] * s0; a.y = arow[ka + 1] * s1;
    b.x = brow[ka];      b.y = brow[ka + 1];
    acc = wmma_f32x4(a, b, acc);
  }
  V8U u; u.v = acc;
  int col = nt * 16 + lm;
  #pragma unroll
  for (int v = 0; v < 8; ++v)
    C[(size_t)(mt * 16 + v + half * 8) * NXC + col] = u.f[v];
}

// ---------------- tiny helpers ----------------
__global__ void k_z0(const float* __restrict__ E, float* __restrict__ z0) {
  int d = threadIdx.x;                                   // 128 threads
  float s = 0.f;
  for (int r = 0; r < NROW; ++r) s += E[(size_t)r * DM + d];
  z0[d] = s * (1.0f / NROW);
}

__global__ void k_rny(const float* __restrict__ Gy, float* __restrict__ rny) {
  int n = blockIdx.x * blockDim.x + threadIdx.x;
  if (n < NYC) rny[n] = rsqrtf(Gy[(size_t)n * NYC + n]);
}

__global__ void k_pairs(int* __restrict__ pi_, int* __restrict__ pk_) {
  int t = blockIdx.x * blockDim.x + threadIdx.x;
  if (t >= NXC * NXC) return;
  int i = t / NXC, k = t % NXC;
  if (i < k) {
    int p = i * (2 * NXC - i - 1) / 2 + (k - i - 1);     // triu_indices order
    pi_[p] = i; pk_[p] = k;
  }
}

__global__ void k_init(float* __restrict__ cx, float* __restrict__ cs,
                       float* __restrict__ total, unsigned* __restrict__ gmax) {
  int t = blockIdx.x * blockDim.x + threadIdx.x;
  if (t < NXC) cx[t] = 0.f;
  if (t < NYC) cs[t] = 0.f;
  if (t == 0) { *total = 0.f; *gmax = 0u; }              // 0 < encf(any float)
}

// ---------------- stage 1 / 3: self-pair softmax + weighted combine + FFN (single block, 1024 thr) ----------------
__global__ void k_selfpair(const float* __restrict__ Q, const float* __restrict__ G,
                           const float* __restrict__ E, const float* __restrict__ z_in,
                           const float* __restrict__ ffnW, const float* __restrict__ ffnb,
                           float* __restrict__ z_out) {
  __shared__ float red[1024];
  __shared__ float cx[NXC];
  __shared__ float st[2];
  __shared__ float rvec[DM];
  const int n = NXC;
  int tid = threadIdx.x;
  float lmax = -3.0e38f;
  for (int t = tid; t < n * n; t += 1024) {
    int i = t / n, k = t % n;
    if (i < k) {
      float s = Q[(size_t)i * n + k];
      float c = G[(size_t)i * n + k] * rsqrtf(G[(size_t)i * n + i] * G[(size_t)k * n + k]);
      if (c < ALPHAC) s = 0.0f;
      lmax = fmaxf(lmax, s);
    }
  }
  red[tid] = lmax; __syncthreads();
  for (int o = 512; o > 0; o >>= 1) { if (tid < o) red[tid] = fmaxf(red[tid], red[tid + o]); __syncthreads(); }
  if (tid == 0) st[0] = red[0];
  for (int j = tid; j < n; j += 1024) cx[j] = 0.0f;
  __syncthreads();
  float gm = st[0], ls = 0.f;
  for (int t = tid; t < n * n; t += 1024) {
    int i = t / n, k = t % n;
    if (i < k) {
      float s = Q[(size_t)i * n + k];
      float c = G[(size_t)i * n + k] * rsqrtf(G[(size_t)i * n + i] * G[(size_t)k * n + k]);
      if (c < ALPHAC) s = 0.0f;
      float w = expf(s - gm);
      ls += w;
      atomicAdd(&cx[i], w);
      atomicAdd(&cx[k], w);
    }
  }
  red[tid] = ls; __syncthreads();
  for (int o = 512; o > 0; o >>= 1) { if (tid < o) red[tid] += red[tid + o]; __syncthreads(); }
  if (tid == 0) st[1] = red[0];
  __syncthreads();
  if (tid < DM) {
    float acc = 0.f;
    for (int j = 0; j < n; ++j) acc += cx[j] * E[(size_t)j * DM + tid];
    rvec[tid] = acc / st[1] + z_in[tid];
  }
  __syncthreads();
  if (tid < DM) {
    float f = ffnb[tid];
    for (int e = 0; e < DM; ++e) f += rvec[e] * ffnW[(size_t)tid * DM + e];
    z_out[tid] = f + rvec[tid];                          // FFN + residual
  }
}

// ---------------- stage 2, pass A: global max over 73536 x 384 virtual score matrix ----------------
// thread p owns score row p; columns in natural order; block-reduce then encoded atomicMax.
__global__ void k_s2max(const float* __restrict__ S2, const float* __restrict__ Gxy,
                        const float* __restrict__ Gx, const float* __restrict__ rny,
                        const int* __restrict__ pi_, const int* __restrict__ pk_,
                        unsigned* __restrict__ gmax) {
  int p = blockIdx.x * blockDim.x + threadIdx.x;
  float lmax = -3.0e38f;
  if (p < NPAIR) {
    int i = pi_[p], k = pk_[p];
    const float* s2i = S2 + (size_t)i * NYC;  const float* s2k = S2 + (size_t)k * NYC;
    const float* gxi = Gxy + (size_t)i * NYC; const float* gxk = Gxy + (size_t)k * NYC;
    float rpn = rsqrtf(Gx[(size_t)i * NXC + i] + 2.0f * Gx[(size_t)i * NXC + k] + Gx[(size_t)k * NXC + k]);
    for (int nn = 0; nn < NYC; ++nn) {
      float s = s2i[nn] + s2k[nn];
      float c = (gxi[nn] + gxk[nn]) * rpn * rny[nn];
      if (c < ALPHAC) s = 0.0f;
      lmax = fmaxf(lmax, s);
    }
  }
  __shared__ float red[256];
  red[threadIdx.x] = lmax; __syncthreads();
  for (int o = 128; o > 0; o >>= 1) { if (threadIdx.x < o) red[threadIdx.x] = fmaxf(red[threadIdx.x], red[threadIdx.x + o]); __syncthreads(); }
  if (threadIdx.x == 0) atomicMax(gmax, encf(red[0]));
}

// ---------------- stage 2, pass B: exp-sum, pair-coeff and column-sum accumulation ----------------
__global__ void k_s2sum(const float* __restrict__ S2, const float* __restrict__ Gxy,
                        const float* __restrict__ Gx, const float* __restrict__ rny,
                        const int* __restrict__ pi_, const int* __restrict__ pk_,
                        const unsigned* __restrict__ gmax,
                        float* __restrict__ cx, float* __restrict__ cs, float* __restrict__ total) {
  __shared__ float cxl[NXC];
  __shared__ float csl[NYC];
  __shared__ float red[256];
  for (int j = threadIdx.x; j < NXC; j += blockDim.x) { cxl[j] = 0.f; csl[j] = 0.f; }
  __syncthreads();
  float gm = decf(*gmax);
  int p = blockIdx.x * blockDim.x + threadIdx.x;
  float lt = 0.f;
  if (p < NPAIR) {
    int i = pi_[p], k = pk_[p];
    const float* s2i = S2 + (size_t)i * NYC;  const float* s2k = S2 + (size_t)k * NYC;
    const float* gxi = Gxy + (size_t)i * NYC; const float* gxk = Gxy + (size_t)k * NYC;
    float rpn = rsqrtf(Gx[(size_t)i * NXC + i] + 2.0f * Gx[(size_t)i * NXC + k] + Gx[(size_t)k * NXC + k]);
    int skew = (threadIdx.x & 31) * 12;                  // 384/32: distinct LDS banks per lane
    float rowsum = 0.f;
    for (int jj = 0; jj < NYC; ++jj) {
      int nn = jj + skew; if (nn >= NYC) nn -= NYC;
      float s = s2i[nn] + s2k[nn];
      float c = (gxi[nn] + gxk[nn]) * rpn * rny[nn];
      if (c < ALPHAC) s = 0.0f;
      float w = expf(s - gm);
      rowsum += w;
      atomicAdd(&csl[nn], w);
    }
    lt = rowsum;
    atomicAdd(&cxl[i], rowsum);
    atomicAdd(&cxl[k], rowsum);
  }
  red[threadIdx.x] = lt; __syncthreads();
  for (int o = 128; o > 0; o >>= 1) { if (threadIdx.x < o) red[threadIdx.x] += red[threadIdx.x + o]; __syncthreads(); }
  if (threadIdx.x == 0) atomicAdd(total, red[0]);
  for (int j = threadIdx.x; j < NXC; j += blockDim.x) {
    if (cxl[j] != 0.f) atomicAdd(&cx[j], cxl[j]);
    if (csl[j] != 0.f) atomicAdd(&cs[j], csl[j]);
  }
}

// ---------------- stage 2, pass C: r = (Σ cx_j ex_j + Σ cs_n ey_n)/total + z; FFN ----------------
__global__ void k_s2combine(const float* __restrict__ cx, const float* __restrict__ cs,
                            const float* __restrict__ total, const float* __restrict__ E,
                            const float* __restrict__ z_in, const float* __restrict__ ffnW,
                            const float* __restrict__ ffnb, float* __restrict__ z_out) {
  __shared__ float rvec[DM];
  int d = threadIdx.x;                                   // 128 threads
  float inv = 1.0f / *total;
  const float* Ex = E;
  const float* Ey = E + (size_t)NXC * DM;
  float acc = 0.f;
  for (int j = 0; j < NXC; ++j) acc += cx[j] * Ex[(size_t)j * DM + d];
  for (int nn = 0; nn < NYC; ++nn) acc += cs[nn] * Ey[(size_t)nn * DM + d];
  rvec[d] = acc * inv + z_in[d];
  __syncthreads();
  float f = ffnb[d];
  for (int e = 0; e < DM; ++e) f += rvec[e] * ffnW[(size_t)d * DM + e];
  z_out[d] = f + rvec[d];
}

// ---------------- QNet head ----------------
__global__ void k_qnet(const float* __restrict__ z_in, const float* __restrict__ W1,
                       const float* __restrict__ b1, const float* __restrict__ W2,
                       const float* __restrict__ b2, float* __restrict__ out) {
  __shared__ float zs[DM];
  __shared__ float h[HIDC];
  int tid = threadIdx.x;                                 // 128 threads
  float zv = z_in[tid]; if (zv != zv) zv = 0.0f;         // _nan0
  zs[tid] = zv; __syncthreads();
  float a = b1[tid];
  for (int d = 0; d < DM; ++d) a += zs[d] * W1[(size_t)tid * DM + d];
  h[tid] = fmaxf(a, 0.0f);
  __syncthreads();
  if (tid < 2) {
    float l = b2[tid];
    for (int j = 0; j < HIDC; ++j) l += h[j] * W2[(size_t)tid * HIDC + j];
    out[tid] = l;                                        // logits
  }
  out[2 + tid] = zs[tid];                                // z
}

extern "C" void kernel_launch(void* const* d_in, const int* in_sizes, int n_in,
                              void* d_out, int out_size, void* d_ws, size_t ws_size,
                              hipStream_t stream) {
  const float* input_x  = (const float*)d_in[0];
  const float* input_y  = (const float*)d_in[1];
  const float* ln_gamma = (const float*)d_in[2];
  const float* ln_beta  = (const float*)d_in[3];
  const float* red_W    = (const float*)d_in[4];
  const float* red_b    = (const float*)d_in[5];
  const float* weight_x = (const float*)d_in[6];
  const float* weight_y = (const float*)d_in[7];
  const float* weight_xy= (const float*)d_in[8];
  const float* ffn_W    = (const float*)d_in[9];
  const float* ffn_b    = (const float*)d_in[10];
  const float* q_W1     = (const float*)d_in[11];
  const float* q_b1     = (const float*)d_in[12];
  const float* q_W2     = (const float*)d_in[13];
  const float* q_b2     = (const float*)d_in[14];
  float* out = (float*)d_out;

  // workspace carve (all 4-byte aligned)
  float* p = (float*)d_ws;
  float* mu    = p; p += NROW;
  float* rstd  = p; p += NROW;
  float* E     = p; p += (size_t)NROW * DM;        // ex rows 0..383, ey rows 384..767
  float* Gx    = p; p += (size_t)NXC * NXC;
  float* Gy    = p; p += (size_t)NYC * NYC;
  float* Gxy   = p; p += (size_t)NXC * NYC;
  float* Qx    = p; p += (size_t)NXC * NXC;
  float* Qy    = p; p += (size_t)NYC * NYC;
  float* S2    = p; p += (size_t)NXC * NYC;
  float* rny   = p; p += NYC;
  float* z0    = p; p += DM;
  float* z1    = p; p += DM;
  float* z2    = p; p += DM;
  float* z3    = p; p += DM;
  float* cx2   = p; p += NXC;
  float* cs2   = p; p += NYC;
  float* total = p; p += 1;
  unsigned* gmax = (unsigned*)p; p += 1;
  int* pi_ = (int*)p; p += NPAIR;
  int* pk_ = (int*)p; p += NPAIR;
  (void)ws_size; (void)n_in; (void)in_sizes; (void)out_size;

  const float* ex = E;
  const float* ey = E + (size_t)NXC * DM;

  k_init  <<<2, 256, 0, stream>>>(cx2, cs2, total, gmax);
  k_pairs <<<(NXC * NXC + 255) / 256, 256, 0, stream>>>(pi_, pk_);
  k_stats <<<NROW, 256, 0, stream>>>(input_x, input_y, mu, rstd);
  // 384 waves (48x8 tiles), 4 waves/block
  k_ln_gemm<<<96, 128, 0, stream>>>(input_x, input_y, mu, rstd,
                                    ln_gamma, ln_beta, red_W, red_b, E);
  // six 384x384x128 WMMA GEMMs: 576 waves each, 4 waves/block
  k_gram<<<144, 128, 0, stream>>>(ex, ex, nullptr,   Gx);
  k_gram<<<144, 128, 0, stream>>>(ey, ey, nullptr,   Gy);
  k_gram<<<144, 128, 0, stream>>>(ex, ey, nullptr,   Gxy);
  k_gram<<<144, 128, 0, stream>>>(ex, ex, weight_x,  Qx);
  k_gram<<<144, 128, 0, stream>>>(ey, ey, weight_y,  Qy);
  k_gram<<<144, 128, 0, stream>>>(ex, ey, weight_xy, S2);

  k_rny<<<2, 256, 0, stream>>>(Gy, rny);
  k_z0 <<<1, DM, 0, stream>>>(E, z0);

  // stage 1
  k_selfpair<<<1, 1024, 0, stream>>>(Qx, Gx, ex, z0, ffn_W, ffn_b, z1);
  // stage 2
  k_s2max<<<(NPAIR + 255) / 256, 256, 0, stream>>>(S2, Gxy, Gx, rny, pi_, pk_, gmax);
  k_s2sum<<<(NPAIR + 255) / 256, 256, 0, stream>>>(S2, Gxy, Gx, rny, pi_, pk_, gmax,
                                                   cx2, cs2, total);
  k_s2combine<<<1, DM, 0, stream>>>(cx2, cs2, total, E, z1, ffn_W, ffn_b, z2);
  // stage 3
  k_selfpair<<<1, 1024, 0, stream>>>(Qy, Gy, ey, z2, ffn_W, ffn_b, z3);
  // head
  k_qnet<<<1, DM, 0, stream>>>(z3, q_W1, q_b1, q_W2, q_b2, out);
}